// FiBiNetPlusLayer_4303557230969
// MI455X (gfx1250) — compile-verified
//
#include <hip/hip_runtime.h>
#include <math.h>

#define B_TOT   16384
#define E       16
#define F       13
#define F_CAT   10
#define F_CONT  3
#define PAIRS   78
#define MID     17
#define EPSV    1e-3f
#define ROWS    16
#define NTHREADS 128
#define XSTR    209      // padded row stride for xs (coprime with 64 banks)
#define PSTR    81       // padded row stride for pt

typedef __attribute__((ext_vector_type(2))) float v2f;
typedef __attribute__((ext_vector_type(8))) float v8f;

// Sum across the 16-lane half-row using DPP (no LDS traffic).
// Valid because values become uniform within each quad before the mirrors.
#define DPP_HALFROW_REDUCE(pr)                                                              \
    do {                                                                                    \
        pr += __int_as_float(__builtin_amdgcn_update_dpp(0, __float_as_int(pr), 0xB1,  0xF, 0xF, true)); \
        pr += __int_as_float(__builtin_amdgcn_update_dpp(0, __float_as_int(pr), 0x4E,  0xF, 0xF, true)); \
        pr += __int_as_float(__builtin_amdgcn_update_dpp(0, __float_as_int(pr), 0x141, 0xF, 0xF, true)); \
        pr += __int_as_float(__builtin_amdgcn_update_dpp(0, __float_as_int(pr), 0x140, 0xF, 0xF, true)); \
    } while (0)

__global__ __launch_bounds__(NTHREADS)
void fibinet_plus_kernel(
    const int*   __restrict__ cat_ids,     // (B,10)
    const int*   __restrict__ cont_keys,   // (B,3)
    const float* __restrict__ cont_values, // (B,3)
    const float* __restrict__ emb,         // (V,16)
    const float* __restrict__ bn_gamma,    // (16)
    const float* __restrict__ bn_beta,     // (16)
    const float* __restrict__ ln_cont_g,   // (3,16)
    const float* __restrict__ ln_cont_b,   // (3,16)
    const float* __restrict__ W_bil,       // (78,16,16)
    const float* __restrict__ bil_dW,      // (78,16)
    const float* __restrict__ bil_db,      // (16)
    const float* __restrict__ bil_ln_g,    // (16)
    const float* __restrict__ bil_ln_b,    // (16)
    const float* __restrict__ se_W1,       // (52,17)
    const float* __restrict__ se_b1,       // (17)
    const float* __restrict__ se_ln1_g,    // (17)
    const float* __restrict__ se_ln1_b,    // (17)
    const float* __restrict__ se_W2,       // (17,208)
    const float* __restrict__ se_b2,       // (208)
    const float* __restrict__ se_ln2_g,    // (208)
    const float* __restrict__ se_ln2_b,    // (208)
    const float* __restrict__ f_W1,        // (224,32)
    const float* __restrict__ f_b1,        // (32)
    const float* __restrict__ f_ln_g,      // (32)
    const float* __restrict__ f_ln_b,      // (32)
    const float* __restrict__ f_W2,        // (32,1)
    const float* __restrict__ f_b2,        // (1)
    float*       __restrict__ out)         // (B,1)
{
    __shared__ float xs[ROWS * XSTR];      // normalized x tile: [row][feat][e], padded
    __shared__ float pt[ROWS * PSTR];      // bilinear pair scalars, padded
    __shared__ float bil_s[ROWS * E];
    __shared__ float info_s[ROWS * 52];
    __shared__ float h_s[ROWS * MID];
    __shared__ float a_s[ROWS * F * E];    // SE pre-act, then v = x * a
    __shared__ float hf_s[ROWS * 32];

    const int tid  = threadIdx.x;
    const int tile = blockIdx.x;

    // ---------------- Phase 0: gather + BN / LN into LDS ----------------
    const float inv_bn = rsqrtf(1.0f + EPSV);
    for (int idx = tid; idx < ROWS * F; idx += NTHREADS) {
        const int r = idx / F, f = idx % F;
        const int gb = tile * ROWS + r;
        float v[E];
        if (f < F_CAT) {
            const int id = cat_ids[gb * F_CAT + f];
            const float4* e4 = (const float4*)(emb + (size_t)id * E);
            #pragma unroll
            for (int q = 0; q < 4; ++q) {
                float4 t = e4[q];
                v[q*4+0] = t.x; v[q*4+1] = t.y; v[q*4+2] = t.z; v[q*4+3] = t.w;
            }
            #pragma unroll
            for (int e = 0; e < E; ++e)
                xs[r*XSTR + f*E + e] = v[e] * inv_bn * bn_gamma[e] + bn_beta[e];
        } else {
            const int cf = f - F_CAT;
            const int id = cont_keys[gb * F_CONT + cf];
            const float cv = cont_values[gb * F_CONT + cf];
            const float4* e4 = (const float4*)(emb + (size_t)id * E);
            #pragma unroll
            for (int q = 0; q < 4; ++q) {
                float4 t = e4[q];
                v[q*4+0] = t.x * cv; v[q*4+1] = t.y * cv;
                v[q*4+2] = t.z * cv; v[q*4+3] = t.w * cv;
            }
            float mu = 0.f;
            #pragma unroll
            for (int e = 0; e < E; ++e) mu += v[e];
            mu *= (1.0f / E);
            float var = 0.f;
            #pragma unroll
            for (int e = 0; e < E; ++e) { float d = v[e] - mu; var += d * d; }
            var *= (1.0f / E);
            const float is = rsqrtf(var + EPSV);
            #pragma unroll
            for (int e = 0; e < E; ++e)
                xs[r*XSTR + f*E + e] =
                    (v[e] - mu) * is * ln_cont_g[cf*E + e] + ln_cont_b[cf*E + e];
        }
    }
    __syncthreads();

    // ---------------- Phase A: bilinear pairs via V_WMMA_F32_16X16X4_F32 ----
    {
        const int wave  = tid >> 5;
        const int lane  = tid & 31;
        const int halfL = lane & 15;
        const int hi    = lane >> 4;          // 0: lanes 0-15, 1: lanes 16-31

        for (int p = wave; p < PAIRS; p += 4) {
            // triu index -> (i,j)
            int i = 0, rem = p;
            while (rem >= (F - 1 - i)) { rem -= (F - 1 - i); ++i; }
            const int j = i + 1 + rem;

            // A operand: Xi tile (batch-row M x K=e). 16x4 fp32 layout per chunk:
            // lanes 0-15 hold K={kb,kb+1}, lanes 16-31 hold K={kb+2,kb+3}.
            const float* xiRow = xs + halfL * XSTR + i * E;
            const int ka = hi * 2;
            v2f a0 = { xiRow[ 0 + ka], xiRow[ 0 + ka + 1] };
            v2f a1 = { xiRow[ 4 + ka], xiRow[ 4 + ka + 1] };
            v2f a2 = { xiRow[ 8 + ka], xiRow[ 8 + ka + 1] };
            v2f a3 = { xiRow[12 + ka], xiRow[12 + ka + 1] };

            // B operand: W_p (K=e rows x N=f cols), mirrored 4x16 layout.
            const float* Wp = W_bil + (size_t)p * (E * E);
            if (p + 4 < PAIRS)
                __builtin_prefetch(W_bil + (size_t)(p + 4) * (E * E), 0, 1);
            const int rk = hi * 2, cn = halfL;
            v2f b0 = { Wp[( 0 + rk)*E + cn], Wp[( 1 + rk)*E + cn] };
            v2f b1 = { Wp[( 4 + rk)*E + cn], Wp[( 5 + rk)*E + cn] };
            v2f b2 = { Wp[( 8 + rk)*E + cn], Wp[( 9 + rk)*E + cn] };
            v2f b3 = { Wp[(12 + rk)*E + cn], Wp[(13 + rk)*E + cn] };

            v8f acc = {};
            acc = __builtin_amdgcn_wmma_f32_16x16x4_f32(false, a0, false, b0, (short)0, acc, false, false);
            acc = __builtin_amdgcn_wmma_f32_16x16x4_f32(false, a1, false, b1, (short)0, acc, false, false);
            acc = __builtin_amdgcn_wmma_f32_16x16x4_f32(false, a2, false, b2, (short)0, acc, false, false);
            acc = __builtin_amdgcn_wmma_f32_16x16x4_f32(false, a3, false, b3, (short)0, acc, false, false);

            // C layout: VGPR r -> M = r + hi*8, N = halfL.
            // p_val[M] = sum_N C[M][N] * xj[M][N]; DPP reduction over the 16-lane half.
            float sums[8];
            #pragma unroll
            for (int r = 0; r < 8; ++r) {
                const int M = r + hi * 8;
                float pr = acc[r] * xs[M * XSTR + j * E + halfL];
                DPP_HALFROW_REDUCE(pr);
                sums[r] = pr;
            }
            if (halfL == 0) {
                #pragma unroll
                for (int r = 0; r < 8; ++r)
                    pt[(r + hi * 8) * PSTR + p] = sums[r];
            }
        }
    }
    __syncthreads();

    // ---------------- Phase B1: SE info (mean/max per half-embedding) -------
    for (int idx = tid; idx < ROWS * F; idx += NTHREADS) {
        const int r = idx / F, f = idx % F;
        const float* base = xs + r * XSTR + f * E;
        float m0 = 0.f, m1 = 0.f, x0 = base[0], x1 = base[8];
        #pragma unroll
        for (int e = 0; e < 8; ++e) {
            m0 += base[e];     x0 = fmaxf(x0, base[e]);
            m1 += base[8 + e]; x1 = fmaxf(x1, base[8 + e]);
        }
        info_s[r * 52 + f * 4 + 0] = m0 * 0.125f;
        info_s[r * 52 + f * 4 + 1] = m1 * 0.125f;
        info_s[r * 52 + f * 4 + 2] = x0;
        info_s[r * 52 + f * 4 + 3] = x1;
    }

    // ---------------- Phase B2: bil = LN(p @ Wd + b) -----------------------
    if (tid < ROWS) {
        const int row = tid;
        float q[E];
        #pragma unroll
        for (int e = 0; e < E; ++e) q[e] = bil_db[e];
        for (int p = 0; p < PAIRS; ++p) {
            const float pv = pt[row * PSTR + p];
            #pragma unroll
            for (int e = 0; e < E; ++e) q[e] += pv * bil_dW[p * E + e];
        }
        float mu = 0.f;
        #pragma unroll
        for (int e = 0; e < E; ++e) mu += q[e];
        mu *= (1.0f / E);
        float var = 0.f;
        #pragma unroll
        for (int e = 0; e < E; ++e) { float d = q[e] - mu; var += d * d; }
        var *= (1.0f / E);
        const float is = rsqrtf(var + EPSV);
        #pragma unroll
        for (int e = 0; e < E; ++e)
            bil_s[row * E + e] = (q[e] - mu) * is * bil_ln_g[e] + bil_ln_b[e];
    }
    __syncthreads();

    // ---------------- Phase B3: h = relu(LN(info @ W1 + b1)) ---------------
    if (tid < ROWS) {
        const int row = tid;
        float t[MID];
        for (int c = 0; c < MID; ++c) {
            float acc = se_b1[c];
            for (int k = 0; k < 52; ++k) acc += info_s[row * 52 + k] * se_W1[k * MID + c];
            t[c] = acc;
        }
        float mu = 0.f;
        for (int c = 0; c < MID; ++c) mu += t[c];
        mu *= (1.0f / MID);
        float var = 0.f;
        for (int c = 0; c < MID; ++c) { float d = t[c] - mu; var += d * d; }
        var *= (1.0f / MID);
        const float is = rsqrtf(var + EPSV);
        for (int c = 0; c < MID; ++c)
            h_s[row * MID + c] =
                fmaxf((t[c] - mu) * is * se_ln1_g[c] + se_ln1_b[c], 0.0f);
    }
    __syncthreads();

    // ---------------- Phase B4: a_pre = h @ W2 + b2 ------------------------
    for (int idx = tid; idx < ROWS * F * E; idx += NTHREADS) {
        const int row = idx / (F * E), c = idx % (F * E);
        float acc = se_b2[c];
        #pragma unroll
        for (int k = 0; k < MID; ++k) acc += h_s[row * MID + k] * se_W2[k * (F * E) + c];
        a_s[idx] = acc;
    }
    __syncthreads();

    // ---------------- Phase B5: a = relu(LN(a_pre)); v = x * a -------------
    if (tid < ROWS) {
        const int row = tid;
        float mu = 0.f;
        for (int c = 0; c < F * E; ++c) mu += a_s[row * F * E + c];
        mu *= (1.0f / (F * E));
        float var = 0.f;
        for (int c = 0; c < F * E; ++c) { float d = a_s[row * F * E + c] - mu; var += d * d; }
        var *= (1.0f / (F * E));
        const float is = rsqrtf(var + EPSV);
        for (int c = 0; c < F * E; ++c) {
            float av = (a_s[row * F * E + c] - mu) * is * se_ln2_g[c] + se_ln2_b[c];
            av = fmaxf(av, 0.0f);
            a_s[row * F * E + c] = xs[row * XSTR + c] * av;   // v
        }
    }
    __syncthreads();

    // ---------------- Phase B6: hf_pre = [bil, v] @ f_W1 + f_b1 ------------
    for (int idx = tid; idx < ROWS * 32; idx += NTHREADS) {
        const int row = idx / 32, c = idx % 32;
        float acc = f_b1[c];
        #pragma unroll
        for (int k = 0; k < E; ++k) acc += bil_s[row * E + k] * f_W1[k * 32 + c];
        for (int k = 0; k < F * E; ++k) acc += a_s[row * F * E + k] * f_W1[(E + k) * 32 + c];
        hf_s[idx] = acc;
    }
    __syncthreads();

    // ---------------- Phase B7: out = sigmoid(relu(LN(hf)) @ f_W2 + f_b2) --
    if (tid < ROWS) {
        const int row = tid;
        float mu = 0.f;
        #pragma unroll
        for (int c = 0; c < 32; ++c) mu += hf_s[row * 32 + c];
        mu *= (1.0f / 32.0f);
        float var = 0.f;
        #pragma unroll
        for (int c = 0; c < 32; ++c) { float d = hf_s[row * 32 + c] - mu; var += d * d; }
        var *= (1.0f / 32.0f);
        const float is = rsqrtf(var + EPSV);
        float o = f_b2[0];
        #pragma unroll
        for (int c = 0; c < 32; ++c) {
            float hv = fmaxf((hf_s[row * 32 + c] - mu) * is * f_ln_g[c] + f_ln_b[c], 0.0f);
            o += hv * f_W2[c];
        }
        out[tile * ROWS + row] = 1.0f / (1.0f + expf(-o));
    }
}

extern "C" void kernel_launch(void* const* d_in, const int* in_sizes, int n_in,
                              void* d_out, int out_size, void* d_ws, size_t ws_size,
                              hipStream_t stream) {
    (void)in_sizes; (void)n_in; (void)d_ws; (void)ws_size; (void)out_size;
    const int*   cat_ids     = (const int*)  d_in[0];
    const int*   cont_keys   = (const int*)  d_in[1];
    const float* cont_values = (const float*)d_in[2];
    const float* emb         = (const float*)d_in[3];
    const float* bn_gamma    = (const float*)d_in[4];
    const float* bn_beta     = (const float*)d_in[5];
    const float* ln_cont_g   = (const float*)d_in[6];
    const float* ln_cont_b   = (const float*)d_in[7];
    const float* W_bil       = (const float*)d_in[8];
    const float* bil_dW      = (const float*)d_in[9];
    const float* bil_db      = (const float*)d_in[10];
    const float* bil_ln_g    = (const float*)d_in[11];
    const float* bil_ln_b    = (const float*)d_in[12];
    const float* se_W1       = (const float*)d_in[13];
    const float* se_b1       = (const float*)d_in[14];
    const float* se_ln1_g    = (const float*)d_in[15];
    const float* se_ln1_b    = (const float*)d_in[16];
    const float* se_W2       = (const float*)d_in[17];
    const float* se_b2       = (const float*)d_in[18];
    const float* se_ln2_g    = (const float*)d_in[19];
    const float* se_ln2_b    = (const float*)d_in[20];
    const float* f_W1        = (const float*)d_in[21];
    const float* f_b1        = (const float*)d_in[22];
    const float* f_ln_g      = (const float*)d_in[23];
    const float* f_ln_b      = (const float*)d_in[24];
    const float* f_W2        = (const float*)d_in[25];
    const float* f_b2        = (const float*)d_in[26];

    dim3 grid(B_TOT / ROWS), block(NTHREADS);
    hipLaunchKernelGGL(fibinet_plus_kernel, grid, block, 0, stream,
        cat_ids, cont_keys, cont_values, emb, bn_gamma, bn_beta,
        ln_cont_g, ln_cont_b, W_bil, bil_dW, bil_db, bil_ln_g, bil_ln_b,
        se_W1, se_b1, se_ln1_g, se_ln1_b, se_W2, se_b2, se_ln2_g, se_ln2_b,
        f_W1, f_b1, f_ln_g, f_ln_b, f_W2, f_b2, (float*)d_out);
}